// VanilaGCN_78597901516829
// MI455X (gfx1250) — compile-verified
//
#include <hip/hip_runtime.h>
#include <hip/hip_bf16.h>
#include <cstdint>
#include <cstddef>

typedef __attribute__((ext_vector_type(2))) float v2f;
typedef __attribute__((ext_vector_type(8))) float v8f;

#define F_DIM   128
#define NCLS    10
#define NGRAPH  64

// ---------------- edge-weight L2 normalization ----------------
__global__ void k_zero1(float* p) { p[0] = 0.f; }

__global__ void k_sumsq(const float* __restrict__ ew, int E, float* __restrict__ sumsq) {
  __shared__ float sm[256];
  int idx = blockIdx.x * 256 + threadIdx.x;
  float v = 0.f;
  if (idx < E) { float w = ew[idx]; v = w * w; }
  sm[threadIdx.x] = v;
  __syncthreads();
  for (int s = 128; s > 0; s >>= 1) {
    if ((int)threadIdx.x < s) sm[threadIdx.x] += sm[threadIdx.x + s];
    __syncthreads();
  }
  if (threadIdx.x == 0) atomicAdd(sumsq, sm[0]);
}

__global__ void k_scale(const float* __restrict__ sumsq, float* __restrict__ scale) {
  scale[0] = 1.f / fmaxf(sqrtf(sumsq[0]), 1e-12f);
}

// ---------------- degree + symmetric norm ----------------
__global__ void k_deg_init(float* __restrict__ deg, int N) {
  int i = blockIdx.x * 256 + threadIdx.x;
  if (i < N) deg[i] = 1.0f;                    // self-loop weight
}

__global__ void k_degree(const int* __restrict__ col, const float* __restrict__ ew,
                         const float* __restrict__ scale, float* __restrict__ deg, int E) {
  int e = blockIdx.x * 256 + threadIdx.x;
  if (e < E) atomicAdd(&deg[col[e]], ew[e] * scale[0]);
}

__global__ void k_dinv(float* __restrict__ deg, int N) {
  int i = blockIdx.x * 256 + threadIdx.x;
  if (i < N) { float d = deg[i]; deg[i] = (d > 0.f) ? rsqrtf(d) : 0.f; }
}

// ---------------- FP32 WMMA GEMM: C[M,128] = A[M,128] @ B[128,128] ----------------
// One wave per 16x16 C tile; K loop of 32 chained V_WMMA_F32_16X16X4_F32.
// ISA 7.12.2 f32 layouts: A 16x4: lanes0-15 K={0,1}, lanes16-31 K={2,3}, M=lane%16.
//                         B 4x16 / C 16x16: N=lane%16, upper half-wave offsets row index.
__global__ __launch_bounds__(32) void k_gemm_wmma(const float* __restrict__ A,
                                                  const float* __restrict__ B,
                                                  float* __restrict__ C, int M) {
  int tile  = blockIdx.x;
  int ntile = tile & 7;          // 128/16 = 8 N tiles
  int mtile = tile >> 3;
  int lane  = threadIdx.x;       // 0..31
  int half  = lane >> 4;         // 0 or 1
  int lr    = lane & 15;
  int m0 = mtile * 16, n0 = ntile * 16;

  int mload = m0 + lr; if (mload > M - 1) mload = M - 1;   // clamp (stores guarded)
  const float* Ar = A + (size_t)mload * F_DIM + 2 * half;  // this lane's A row, K-pair base
  const float* Bc = B + (size_t)(2 * half) * F_DIM + n0 + lr;

  v8f acc = {};
#pragma unroll
  for (int k0 = 0; k0 < F_DIM; k0 += 4) {
    v2f a, b;
    a.x = Ar[k0 + 0];                       // A[m][k0+2h]
    a.y = Ar[k0 + 1];                       // A[m][k0+2h+1]
    b.x = Bc[(size_t)k0 * F_DIM];           // B[k0+2h][n]
    b.y = Bc[(size_t)(k0 + 1) * F_DIM];     // B[k0+2h+1][n]
    acc = __builtin_amdgcn_wmma_f32_16x16x4_f32(false, a, false, b,
                                                (short)0, acc, false, false);
  }
#pragma unroll
  for (int i = 0; i < 8; i++) {
    int m = m0 + i + 8 * half;
    if (m < M) C[(size_t)m * F_DIM + n0 + lr] = acc[i];
  }
}

// ---------------- self-loop init: agg = dinv[i]^2 * xw[i] ----------------
__global__ void k_self_init(const float* __restrict__ dinv, const float* __restrict__ xw,
                            float* __restrict__ agg, size_t total) {
  size_t idx = (size_t)blockIdx.x * 256 + threadIdx.x;
  if (idx < total) {
    float di = dinv[idx >> 7];
    agg[idx] = di * di * xw[idx];
  }
}

// ---------------- edge scatter: one wave per edge, float4 per lane ----------------
__global__ void k_scatter(const int* __restrict__ row, const int* __restrict__ col,
                          const float* __restrict__ ew, const float* __restrict__ scale,
                          const float* __restrict__ dinv, const float* __restrict__ xw,
                          float* __restrict__ agg, int E) {
  int e    = blockIdx.x * (256 / 32) + ((int)threadIdx.x >> 5);
  int lane = threadIdx.x & 31;
  if (e >= E) return;
  int r = row[e], c = col[e];
  float norm = dinv[r] * (ew[e] * scale[0]) * dinv[c];
  const float4* src = (const float4*)(xw + (size_t)r * F_DIM);
  float4 v = src[lane];                               // 32 lanes x 4 = 128 feats
  float* dst = agg + (size_t)c * F_DIM + lane * 4;
  atomicAdd(dst + 0, norm * v.x);
  atomicAdd(dst + 1, norm * v.y);
  atomicAdd(dst + 2, norm * v.z);
  atomicAdd(dst + 3, norm * v.w);
}

// ---------------- bias + ReLU ----------------
__global__ void k_bias_relu(const float* __restrict__ agg, const float* __restrict__ b,
                            float* __restrict__ h, size_t total) {
  size_t idx = (size_t)blockIdx.x * 256 + threadIdx.x;
  if (idx < total) h[idx] = fmaxf(agg[idx] + b[idx & 127], 0.f);
}

// ---------------- segment-max pool (exact: post-ReLU values are >= 0) ----------------
__global__ void k_pool_init(unsigned* __restrict__ p) {
  p[blockIdx.x * 256 + threadIdx.x] = 0u;             // bits of 0.0f
}

__global__ void k_pool(const float* __restrict__ h, const int* __restrict__ batch,
                       unsigned* __restrict__ pool, size_t total) {
  size_t idx = (size_t)blockIdx.x * 256 + threadIdx.x;
  if (idx < total) {
    int i = (int)(idx >> 7), f = (int)(idx & 127);
    int g = batch[i];
    atomicMax(&pool[(size_t)g * F_DIM + f], __float_as_uint(h[idx]));
  }
}

// ---------------- final linear: [64,128] @ [128,10] + b ----------------
__global__ void k_final(const unsigned* __restrict__ pool, const float* __restrict__ Wl,
                        const float* __restrict__ bl, float* __restrict__ out) {
  int t = blockIdx.x * 64 + threadIdx.x;
  if (t >= NGRAPH * NCLS) return;
  int g = t / NCLS, c = t % NCLS;
  float s = bl[c];
#pragma unroll 4
  for (int f = 0; f < F_DIM; f++)
    s += __uint_as_float(pool[(size_t)g * F_DIM + f]) * Wl[f * NCLS + c];
  out[t] = s;
}

extern "C" void kernel_launch(void* const* d_in, const int* in_sizes, int n_in,
                              void* d_out, int out_size, void* d_ws, size_t ws_size,
                              hipStream_t stream) {
  (void)n_in; (void)out_size; (void)ws_size;
  const float* x     = (const float*)d_in[0];
  const int*   ei    = (const int*)  d_in[1];
  const float* ew    = (const float*)d_in[2];
  const int*   batch = (const int*)  d_in[3];
  const float* W1 = (const float*)d_in[4];  const float* b1 = (const float*)d_in[5];
  const float* W2 = (const float*)d_in[6];  const float* b2 = (const float*)d_in[7];
  const float* W3 = (const float*)d_in[8];  const float* b3 = (const float*)d_in[9];
  const float* Wl = (const float*)d_in[10]; const float* bl = (const float*)d_in[11];

  const int N = in_sizes[0] / F_DIM;   // 50000
  const int E = in_sizes[1] / 2;       // 800000
  const int* row = ei;
  const int* col = ei + E;

  // workspace layout (floats): [0]=sumsq, [1]=scale, deg@128, then 3 NxF buffers, pool bits
  float*  ws    = (float*)d_ws;
  float*  sumsq = ws;
  float*  scale = ws + 1;
  float*  deg   = ws + 128;                       // doubles as dinv (in place)
  size_t  nf    = (size_t)N * F_DIM;
  size_t  degPad = ((size_t)N + 127) & ~(size_t)127;
  float*  xw    = ws + 128 + degPad;
  float*  agg   = xw + nf;
  float*  h     = agg + nf;
  unsigned* pool = (unsigned*)(h + nf);

  const int nBlk  = (N + 255) / 256;
  const int eBlk  = (E + 255) / 256;
  const int nfBlk = (int)((nf + 255) / 256);
  const int scBlk = (E + 7) / 8;                  // 8 waves (edges) per 256-thread block
  const int gemmGrid = ((N + 15) / 16) * (F_DIM / 16);

  // edge-weight L2 normalization scalar
  k_zero1   <<<1, 1, 0, stream>>>(sumsq);
  k_sumsq   <<<eBlk, 256, 0, stream>>>(ew, E, sumsq);
  k_scale   <<<1, 1, 0, stream>>>(sumsq, scale);

  // degrees with self-loops -> dinv
  k_deg_init<<<nBlk, 256, 0, stream>>>(deg, N);
  k_degree  <<<eBlk, 256, 0, stream>>>(col, ew, scale, deg, E);
  k_dinv    <<<nBlk, 256, 0, stream>>>(deg, N);

  const float* Wmat[3] = {W1, W2, W3};
  const float* bvec[3] = {b1, b2, b3};
  const float* in = x;
  for (int l = 0; l < 3; l++) {
    k_gemm_wmma<<<gemmGrid, 32, 0, stream>>>(in, Wmat[l], xw, N);
    k_self_init<<<nfBlk, 256, 0, stream>>>(deg, xw, agg, nf);
    k_scatter  <<<scBlk, 256, 0, stream>>>(row, col, ew, scale, deg, xw, agg, E);
    k_bias_relu<<<nfBlk, 256, 0, stream>>>(agg, bvec[l], h, nf);
    in = h;
  }

  k_pool_init<<<(NGRAPH * F_DIM) / 256, 256, 0, stream>>>(pool);
  k_pool     <<<nfBlk, 256, 0, stream>>>(h, batch, pool, nf);
  k_final    <<<(NGRAPH * NCLS + 63) / 64, 64, 0, stream>>>(pool, Wl, bl, (float*)d_out);
}